// AttnDecoder_7954279432562
// MI455X (gfx1250) — compile-verified
//
#include <hip/hip_runtime.h>
#include <hip/hip_bf16.h>
#include <math.h>

// Problem dimensions (from the reference)
#define DHID   1024          // INPUT_SIZE
#define SEQ    32768         // SEQ_LEN
#define VOCAB  50257         // OUTPUT_SIZE
#define DCAT   2048          // 2 * INPUT_SIZE
#define RGRP   128           // row-groups for deterministic attn reduction

typedef __attribute__((ext_vector_type(2))) float v2f;
typedef __attribute__((ext_vector_type(8))) float v8f;

// ---------------------------------------------------------------------------
// K1: x = relu(input @ W_h^T + b_h)   (1024 rows, 1024-long dot each)
// one wave per output row, float4 loads, shuffle reduction
// ---------------------------------------------------------------------------
__global__ void __launch_bounds__(256) hidden_kernel(
    const float* __restrict__ inp, const float* __restrict__ Wh,
    const float* __restrict__ bh, float* __restrict__ x)
{
    const int t    = threadIdx.x;
    const int lane = t & 31;
    const int wave = t >> 5;
    const int row  = blockIdx.x * 8 + wave;      // grid = 128 blocks
    const float4* wr = (const float4*)(Wh + (size_t)row * DHID);
    const float4* ir = (const float4*)inp;
    float s = 0.f;
    #pragma unroll
    for (int i = 0; i < DHID / 4; i += 32) {
        float4 w = wr[i + lane];
        float4 u = ir[i + lane];
        s += w.x * u.x + w.y * u.y + w.z * u.z + w.w * u.w;
    }
    for (int off = 16; off > 0; off >>= 1) s += __shfl_down(s, off);
    if (lane == 0) x[row] = fmaxf(s + bh[row], 0.f);
}

// ---------------------------------------------------------------------------
// K2: scores[r] = enc[r,:] . q   via V_WMMA_F32_16X16X4_F32
// Each wave computes 16 rows. A = 16x4 tile of enc (per-lane b64 load),
// B = q chunk broadcast across all 16 columns (b64 from LDS).
// A layout (ISA 7.12.2): lane<16 -> M=lane, VGPRs K={0,1}; lane>=16 -> K={2,3}.
// C/D layout: VGPR v holds M=v (lanes 0-15) / M=v+8 (lanes 16-31).
// ---------------------------------------------------------------------------
__global__ void __launch_bounds__(256) scores_wmma_kernel(
    const float* __restrict__ enc, const float* __restrict__ x,
    float* __restrict__ scores)
{
    __shared__ __align__(16) float sq[DHID];
    const int t = threadIdx.x;
    for (int i = t; i < DHID; i += 256) sq[i] = x[i];
    __syncthreads();

    const int lane  = t & 31;
    const int wave  = t >> 5;
    const int r0    = (blockIdx.x * 8 + wave) * 16;   // grid = 256 blocks
    const int m     = lane & 15;
    const int khalf = lane >> 4;                      // k offset 0 or 2 (in pairs)

    const float2* arow = (const float2*)(enc + (size_t)(r0 + m) * DHID);
    const float2* q2   = (const float2*)sq;

    v8f c = {};
    for (int k0 = 0; k0 < DHID; k0 += 4) {
        const int idx2 = (k0 >> 1) + khalf;
        float2 af = arow[idx2];
        float2 bf = q2[idx2];
        v2f a; a[0] = af.x; a[1] = af.y;
        v2f b; b[0] = bf.x; b[1] = bf.y;
        c = __builtin_amdgcn_wmma_f32_16x16x4_f32(false, a, false, b,
                                                  (short)0, c, false, false);
    }
    if (lane == 0) {
        #pragma unroll
        for (int v = 0; v < 8; ++v) scores[r0 + v] = c[v];
    } else if (lane == 16) {
        #pragma unroll
        for (int v = 0; v < 8; ++v) scores[r0 + 8 + v] = c[v];
    }
}

// ---------------------------------------------------------------------------
// K3: softmax prep over scores: m = max, scores[i] <- exp(scores[i]-m),
// stat[1] = sum(exp). Single block, deterministic tree reductions.
// ---------------------------------------------------------------------------
__global__ void __launch_bounds__(1024) softmax_scores_kernel(
    float* __restrict__ scores, float* __restrict__ stat)
{
    __shared__ float red[1024];
    const int t = threadIdx.x;
    float lm = -3.402823466e38f;
    for (int i = t; i < SEQ; i += 1024) lm = fmaxf(lm, scores[i]);
    red[t] = lm; __syncthreads();
    for (int off = 512; off > 0; off >>= 1) {
        if (t < off) red[t] = fmaxf(red[t], red[t + off]);
        __syncthreads();
    }
    const float mx = red[0];
    __syncthreads();
    float ls = 0.f;
    for (int i = t; i < SEQ; i += 1024) {
        float e = __expf(scores[i] - mx);
        scores[i] = e;
        ls += e;
    }
    red[t] = ls; __syncthreads();
    for (int off = 512; off > 0; off >>= 1) {
        if (t < off) red[t] += red[t + off];
        __syncthreads();
    }
    if (t == 0) stat[1] = red[0];
}

// ---------------------------------------------------------------------------
// K4a: partial[g, c] = sum over rows of group g of e[r] * enc[r, c]
// (second 128 MB pass; hits L2). Each thread owns 4 consecutive columns
// (float4), block owns a 256-row slab. Fully deterministic (no atomics).
// ---------------------------------------------------------------------------
__global__ void __launch_bounds__(256) attn_partial_kernel(
    const float* __restrict__ enc, const float* __restrict__ escore,
    float* __restrict__ part)
{
    const int t = threadIdx.x;
    const int g = blockIdx.x;                 // 0..RGRP-1
    const int rbeg = g * (SEQ / RGRP);
    const int rend = rbeg + (SEQ / RGRP);
    float4 acc = make_float4(0.f, 0.f, 0.f, 0.f);
    for (int r = rbeg; r < rend; ++r) {
        const float e = escore[r];
        float4 v = ((const float4*)(enc + (size_t)r * DHID))[t];
        acc.x += e * v.x; acc.y += e * v.y;
        acc.z += e * v.z; acc.w += e * v.w;
    }
    ((float4*)(part + (size_t)g * DHID))[t] = acc;
}

// K4b: attn[c] = sum_g part[g, c]   (deterministic fixed-order sum)
__global__ void __launch_bounds__(256) attn_reduce_kernel(
    const float* __restrict__ part, float* __restrict__ attn)
{
    const int c = blockIdx.x * 256 + threadIdx.x;   // grid = 4 blocks
    float s = 0.f;
    for (int g = 0; g < RGRP; ++g) s += part[(size_t)g * DHID + c];
    attn[c] = s;
}

// ---------------------------------------------------------------------------
// K5: logits[v] = W_out[v,:] . concat + b_out[v]   via WMMA f32 16x16x4
// concat = [x, attn/sum] staged in LDS. 412 MB stream -> the dominant kernel.
// ---------------------------------------------------------------------------
__global__ void __launch_bounds__(256) logits_wmma_kernel(
    const float* __restrict__ Wout, const float* __restrict__ bout,
    const float* __restrict__ x, const float* __restrict__ attn,
    const float* __restrict__ stat, float* __restrict__ logits)
{
    __shared__ __align__(16) float sc[DCAT];
    const int t = threadIdx.x;
    const float invS = 1.0f / stat[1];
    for (int i = t; i < DCAT; i += 256)
        sc[i] = (i < DHID) ? x[i] : attn[i - DHID] * invS;
    __syncthreads();

    const int lane  = t & 31;
    const int wave  = t >> 5;
    const int r0    = (blockIdx.x * 8 + wave) * 16;   // grid = 393 blocks
    const int m     = lane & 15;
    const int khalf = lane >> 4;

    int mrow = r0 + m;
    if (mrow > VOCAB - 1) mrow = VOCAB - 1;           // clamp tail loads
    const float2* arow = (const float2*)(Wout + (size_t)mrow * DCAT);
    const float2* q2   = (const float2*)sc;

    v8f c = {};
    for (int k0 = 0; k0 < DCAT; k0 += 4) {
        const int idx2 = (k0 >> 1) + khalf;
        float2 af = arow[idx2];
        float2 bf = q2[idx2];
        v2f a; a[0] = af.x; a[1] = af.y;
        v2f b; b[0] = bf.x; b[1] = bf.y;
        c = __builtin_amdgcn_wmma_f32_16x16x4_f32(false, a, false, b,
                                                  (short)0, c, false, false);
    }
    if (lane == 0) {
        #pragma unroll
        for (int v = 0; v < 8; ++v) {
            int row = r0 + v;
            if (row < VOCAB) logits[row] = c[v] + bout[row];
        }
    } else if (lane == 16) {
        #pragma unroll
        for (int v = 0; v < 8; ++v) {
            int row = r0 + 8 + v;
            if (row < VOCAB) logits[row] = c[v] + bout[row];
        }
    }
}

// K6: softmax prep over logits (max, in-place exp, stat[3] = sum)
__global__ void __launch_bounds__(1024) softmax_logits_kernel(
    float* __restrict__ logits, float* __restrict__ stat)
{
    __shared__ float red[1024];
    const int t = threadIdx.x;
    float lm = -3.402823466e38f;
    for (int i = t; i < VOCAB; i += 1024) lm = fmaxf(lm, logits[i]);
    red[t] = lm; __syncthreads();
    for (int off = 512; off > 0; off >>= 1) {
        if (t < off) red[t] = fmaxf(red[t], red[t + off]);
        __syncthreads();
    }
    const float mx = red[0];
    __syncthreads();
    float ls = 0.f;
    for (int i = t; i < VOCAB; i += 1024) {
        float e = __expf(logits[i] - mx);
        logits[i] = e;
        ls += e;
    }
    red[t] = ls; __syncthreads();
    for (int off = 512; off > 0; off >>= 1) {
        if (t < off) red[t] += red[t + off];
        __syncthreads();
    }
    if (t == 0) stat[3] = red[0];
}

// K7: out[v] = e[v] / sum
__global__ void __launch_bounds__(256) normalize_kernel(
    const float* __restrict__ logits, const float* __restrict__ stat,
    float* __restrict__ out)
{
    const int i = blockIdx.x * 256 + threadIdx.x;
    if (i < VOCAB) out[i] = logits[i] * (1.0f / stat[3]);
}

// ---------------------------------------------------------------------------
// Launch
// ---------------------------------------------------------------------------
extern "C" void kernel_launch(void* const* d_in, const int* in_sizes, int n_in,
                              void* d_out, int out_size, void* d_ws, size_t ws_size,
                              hipStream_t stream) {
    (void)in_sizes; (void)n_in; (void)out_size; (void)ws_size;
    const float* inp  = (const float*)d_in[0];   // [1,1,1024]
    const float* enc  = (const float*)d_in[1];   // [32768,1024]
    const float* Wh   = (const float*)d_in[2];   // [1024,1024]
    const float* bh   = (const float*)d_in[3];   // [1024]
    const float* Wout = (const float*)d_in[4];   // [50257,2048]
    const float* bout = (const float*)d_in[5];   // [50257]
    float* out = (float*)d_out;                  // [50257]

    // workspace layout (floats)
    float* ws     = (float*)d_ws;
    float* x      = ws;                               // 1024
    float* scores = x + DHID;                         // 32768
    float* attn   = scores + SEQ;                     // 1024
    float* logits = attn + DHID;                      // 50257
    float* part   = ws + 85088;                       // 128*1024 (16B aligned)
    float* stat   = part + (size_t)RGRP * DHID;       // 4

    hidden_kernel        <<<DHID / 8,              256,  0, stream>>>(inp, Wh, bh, x);
    scores_wmma_kernel   <<<SEQ / 128,             256,  0, stream>>>(enc, x, scores);
    softmax_scores_kernel<<<1,                     1024, 0, stream>>>(scores, stat);
    attn_partial_kernel  <<<RGRP,                  256,  0, stream>>>(enc, scores, part);
    attn_reduce_kernel   <<<DHID / 256,            256,  0, stream>>>(part, attn);
    logits_wmma_kernel   <<<(VOCAB + 127) / 128,   256,  0, stream>>>(Wout, bout, x, attn, stat, logits);
    softmax_logits_kernel<<<1,                     1024, 0, stream>>>(logits, stat);
    normalize_kernel     <<<(VOCAB + 255) / 256,   256,  0, stream>>>(logits, stat, out);
}